// DiscreteWaveletTransform_58385785422123
// MI455X (gfx1250) — compile-verified
//
#include <hip/hip_runtime.h>

typedef float v2f __attribute__((ext_vector_type(2)));
typedef float v4f __attribute__((ext_vector_type(4)));
typedef float v8f __attribute__((ext_vector_type(8)));

#define T_LEN    2048
#define N_COLS   512
#define NROW     16
#define NTHREADS 256
#define NWAVES   8

// Level geometry: S_out = (S_in + 7) / 2  (db4, F=8)
// All LDS row strides are == 18 (mod 64) and == 2 (mod 4):
//  - even => 8-byte aligned v2f (ds_load_b64) fragments
//  - nn*stride mod 64 distinct for nn in [0,16)  => conflict-free across rows
//  - 18*D == 2 (mod 64) has no solution D<16     => half-wave +2 col offset conflict-free
#define S_IN0   2048
#define ST_SIG  2066
#define S_OUT1  1027
#define TILES1  65            // ceil(1027/16); writes reach col 1039
#define ST_1    1042
#define S_OUT2  517
#define TILES2  33            // writes reach col 527
#define ST_2    530
#define S_OUT3  262
#define TILES3  17            // writes reach col 271
#define ST_3    274

// LDS float offsets (sig region is reused for level-2/3 buffers)
#define OFF_SIG 0
#define OFF_A1  (NROW * ST_SIG)          // 33056
#define OFF_D1  (OFF_A1 + NROW * ST_1)   // 49728
#define OFF_A2  0
#define OFF_D2  (NROW * ST_2)            // 8480
#define OFF_A3  (2 * NROW * ST_2)        // 16960
#define OFF_D3  (OFF_A3 + NROW * ST_3)   // 21344 (ends 25728 < 33056: fits in sig region)
#define LDS_FLOATS (OFF_D1 + NROW * ST_1) // 66400 floats = 265600 B (< 320 KB/WGP)

__device__ __constant__ float c_dec_lo[8] = {
    -0.010597401784997278f, 0.032883011666982945f, 0.030841381835986965f,
    -0.18703481171888114f, -0.02798376941698385f, 0.6308807679295904f,
    0.7148465705525415f, 0.23037781330885523f };
__device__ __constant__ float c_dec_hi[8] = {
    -0.23037781330885523f, 0.7148465705525415f, -0.6308807679295904f,
    -0.02798376941698385f, 0.18703481171888114f, 0.030841381835986965f,
    -0.032883011666982945f, -0.010597401784997278f };

// pywt 'symmetric' extension with leading sample dropped:
// ext[j] = x[mapSym(j, S)], valid for j in [0, 2S+5]; result in [0, S-1].
__device__ __forceinline__ int mapSym(int j, int S) {
    int idx = j - 6;
    if (j < 6)      idx = 5 - j;
    if (j >= S + 6) idx = 2 * S + 5 - j;
    return idx;
}

__device__ __forceinline__ float tapf(const float* h, int s) {
    return (s >= 0 && s < 8) ? h[s] : 0.0f;
}

// Linear resize (torch align_corners=False), S < T case only.
__device__ __forceinline__ float rsz(const float* c, int S, int tt) {
    float pos = ((float)tt + 0.5f) * ((float)S * (1.0f / 2048.0f)) - 0.5f;
    pos = fminf(fmaxf(pos, 0.0f), (float)(S - 1));
    int lo = (int)pos;
    int hi = (lo + 1 < S) ? lo + 1 : S - 1;
    float w = pos - (float)lo;
    return c[lo] + (c[hi] - c[lo]) * w;
}

// One DWT level for 16 rows via banded-matrix f32 WMMA.
// y[f][o] = sum_s h_f[s] * ext[2o + 7 - s]
// D[m][n] = sum_{c,k} A_c[m][k] * B_c[k][n], A_c[m][k] = ext[m][2*n0 + 4c + k],
//           B_c[k][n] = h[2n + 7 - 4c - k] (zero outside [0,7]); K spans 0..39.
__device__ __forceinline__ void dwt_level(float* lds, int srcOff, int srcStride, int Sin,
                                          int dstA, int dstD, int dstStride, int tiles,
                                          int wave, int nn, int half,
                                          const v2f* bLo, const v2f* bHi) {
    const int srcRow = srcOff + nn * srcStride; // A frag: lane<16 -> K0/K1, lane>=16 -> K2/K3, row = lane&15
    for (int tile = wave; tile < tiles; tile += NWAVES) {
        const int n0 = tile * 16;
        v2f a[10];
        // Interior tiles: identity index map for all j in [2n0, 2n0+39] -> contiguous b64 loads.
        if (n0 >= 3 && 2 * n0 + 34 <= Sin) {
            const float* p = &lds[srcRow + 2 * n0 + 2 * half - 6]; // even float idx -> 8B aligned
#pragma unroll
            for (int c = 0; c < 10; ++c)
                a[c] = *(const v2f*)(p + 4 * c);
        } else {
#pragma unroll
            for (int c = 0; c < 10; ++c) {
                const int jb = 2 * n0 + 4 * c + 2 * half;
                a[c].x = lds[srcRow + mapSym(jb,     Sin)];
                a[c].y = lds[srcRow + mapSym(jb + 1, Sin)];
            }
        }
        v8f accA = {0.f, 0.f, 0.f, 0.f, 0.f, 0.f, 0.f, 0.f};
        v8f accD = {0.f, 0.f, 0.f, 0.f, 0.f, 0.f, 0.f, 0.f};
#pragma unroll
        for (int c = 0; c < 10; ++c) {
            accA = __builtin_amdgcn_wmma_f32_16x16x4_f32(false, a[c], false, bLo[c],
                                                         (short)0, accA, false, false);
            accD = __builtin_amdgcn_wmma_f32_16x16x4_f32(false, a[c], false, bHi[c],
                                                         (short)0, accD, false, false);
        }
        const int col = n0 + nn; // C/D layout: VGPR r, lanes<16 -> M=r, lanes>=16 -> M=r+8; N = lane&15
#pragma unroll
        for (int r = 0; r < 8; ++r) {
            const int row = r + 8 * half;
            lds[dstA + row * dstStride + col] = accA[r];
            lds[dstD + row * dstStride + col] = accD[r];
        }
    }
}

extern "C" __global__ void __launch_bounds__(NTHREADS)
dwt_db4_l3_kernel(const float* __restrict__ x, float* __restrict__ out) {
    extern __shared__ float lds[];
    const int tid  = threadIdx.x;
    const int lane = tid & 31;
    const int wave = tid >> 5;
    const int nn   = lane & 15;
    const int half = lane >> 4;

    const int wg  = blockIdx.x;         // 512 workgroups: 16 b * 32 n-tiles
    const int b   = wg >> 5;
    const int n0g = (wg & 31) * NROW;

    // Banded-filter B fragments (depend only on lane & chunk).
    // B layout mirrors C/D striping: VGPR v holds K=v (lanes<16), K=v+2 (lanes>=16).
    v2f bLo[10], bHi[10];
#pragma unroll
    for (int c = 0; c < 10; ++c) {
        const int sA = 2 * nn + 7 - 4 * c - 2 * half;
        bLo[c].x = tapf(c_dec_lo, sA);     bLo[c].y = tapf(c_dec_lo, sA - 1);
        bHi[c].x = tapf(c_dec_hi, sA);     bHi[c].y = tapf(c_dec_hi, sA - 1);
    }

    // Stage input tile into LDS, transposing [t][n] -> sig[n][t]. Read-once: NT loads.
    const float* gx = x + ((size_t)b * T_LEN) * N_COLS + n0g;
    {
        const int nl = tid & 15;
        const int ts = tid >> 4;          // 16 t-values per iteration
#pragma unroll 4
        for (int it = 0; it < T_LEN / 16; ++it) {
            const int t = it * 16 + ts;
            lds[OFF_SIG + nl * ST_SIG + t] =
                __builtin_nontemporal_load(&gx[(size_t)t * N_COLS + nl]);
        }
    }
    __syncthreads();

    dwt_level(lds, OFF_SIG, ST_SIG, S_IN0,  OFF_A1, OFF_D1, ST_1, TILES1, wave, nn, half, bLo, bHi);
    __syncthreads();
    dwt_level(lds, OFF_A1,  ST_1,   S_OUT1, OFF_A2, OFF_D2, ST_2, TILES2, wave, nn, half, bLo, bHi);
    __syncthreads();
    dwt_level(lds, OFF_A2,  ST_2,   S_OUT2, OFF_A3, OFF_D3, ST_3, TILES3, wave, nn, half, bLo, bHi);
    __syncthreads();

    // Resize all 4 bands to T; stream out with NT 128-bit stores (write-once 256 MB > L2).
    float* gout = out + (((size_t)b * N_COLS + n0g) * T_LEN) * 4;
    for (int idx = tid; idx < NROW * T_LEN; idx += NTHREADS) {
        const int row = idx >> 11;
        const int tt  = idx & (T_LEN - 1);
        v4f o;
        o.x = rsz(&lds[OFF_A3 + row * ST_3], S_OUT3, tt);  // cA3
        o.y = rsz(&lds[OFF_D3 + row * ST_3], S_OUT3, tt);  // cD3
        o.z = rsz(&lds[OFF_D2 + row * ST_2], S_OUT2, tt);  // cD2
        o.w = rsz(&lds[OFF_D1 + row * ST_1], S_OUT1, tt);  // cD1
        __builtin_nontemporal_store(o, (v4f*)&gout[((size_t)row * T_LEN + tt) * 4]);
    }
}

extern "C" void kernel_launch(void* const* d_in, const int* in_sizes, int n_in,
                              void* d_out, int out_size, void* d_ws, size_t ws_size,
                              hipStream_t stream) {
    const float* x = (const float*)d_in[0];
    float* out = (float*)d_out;
    const size_t shmem = (size_t)LDS_FLOATS * sizeof(float); // 265600 B
    (void)hipFuncSetAttribute(reinterpret_cast<const void*>(dwt_db4_l3_kernel),
                              hipFuncAttributeMaxDynamicSharedMemorySize, (int)shmem);
    dwt_db4_l3_kernel<<<dim3(512), dim3(NTHREADS), shmem, stream>>>(x, out);
}